// GatedDeltaNetBlock_4475355922510
// MI455X (gfx1250) — compile-verified
//
#include <hip/hip_runtime.h>

// Gated DeltaNet chunkwise recurrence for gfx1250 (MI455X), f32 WMMA path.
// B=2, H=16, S=4096, Dk=128, Dv=256, CHUNK=64.
// Grid: (B*H) * (Dv/64) = 128 blocks, 256 threads (8 wave32s) each.
// v-tile is fetched by the Tensor Data Mover (tensor_load_to_lds) when available.

typedef float v2f __attribute__((ext_vector_type(2)));
typedef float v8f __attribute__((ext_vector_type(8)));
typedef unsigned int u32x4 __attribute__((ext_vector_type(4)));
typedef int i32x4 __attribute__((ext_vector_type(4)));
typedef int i32x8 __attribute__((ext_vector_type(8)));

#define LQ    64
#define DKk   128
#define DVv   256
#define DVB   64
#define NCH   64
#define SEQ   4096
#define QSCALE 0.08838834764831845f   // 128^-0.5
#define EPSN  1e-6f

#define ZACC {0, 0, 0, 0, 0, 0, 0, 0}

#if defined(__has_builtin)
#if __has_builtin(__builtin_amdgcn_tensor_load_to_lds) && __has_builtin(__builtin_amdgcn_s_wait_tensorcnt)
#define HAVE_TDM 1
#endif
#endif

#define WMMA4(c, a, b) \
  (c) = __builtin_amdgcn_wmma_f32_16x16x4_f32(false, (a), false, (b), (short)0, (c), false, false)

// Fragment loaders with precomputed per-lane base pointers (immediate-friendly).
struct Contig {             // elements at p[kk], p[kk+1]  (A-frags and B^T-frags)
  const float* p;
  __device__ __forceinline__ v2f operator()(int kk) const {
    v2f r; r.x = p[kk]; r.y = p[kk + 1]; return r;
  }
};
struct Strided {            // elements at p[kk*ld], p[kk*ld+ld]  (row-major B-frags)
  const float* p; int ld;
  __device__ __forceinline__ v2f operator()(int kk) const {
    v2f r; r.x = p[kk * ld]; r.y = p[kk * ld + ld]; return r;
  }
};
struct CorrB {              // beta_k * (eg_k * v[k][n] - Y[k][n])
  const float* pv; const float* py; const float* bt; const float* eg;
  __device__ __forceinline__ v2f operator()(int kk) const {
    v2f r;
    r.x = bt[kk]     * (eg[kk]     * pv[kk * DVB]       - py[kk * LQ]);
    r.y = bt[kk + 1] * (eg[kk + 1] * pv[kk * DVB + DVB] - py[kk * LQ + LQ]);
    return r;
  }
};

// Software-pipelined 16x32 tile matmul: 1 shared A-frag, 2 B-frags per K step.
template <typename AL, typename B0, typename B1>
static __device__ __forceinline__ void mm2p(v8f& a0, v8f& a1, int K, AL al, B0 b0l, B1 b1l) {
  v2f a = al(0);
  v2f b0 = b0l(0);
  v2f b1 = b1l(0);
#pragma unroll 4
  for (int kk = 4; kk < K; kk += 4) {
    v2f an = al(kk);
    v2f b0n = b0l(kk);
    v2f b1n = b1l(kk);
    WMMA4(a0, a, b0);
    WMMA4(a1, a, b1);
    a = an; b0 = b0n; b1 = b1n;
  }
  WMMA4(a0, a, b0);
  WMMA4(a1, a, b1);
}

// Software-pipelined 16x64 tile matmul: 1 shared A-frag, 4 B-frags per K step.
template <typename AL, typename BL>
static __device__ __forceinline__ void mm4p(v8f& a0, v8f& a1, v8f& a2, v8f& a3,
                                            int K, AL al, BL b0l, BL b1l, BL b2l, BL b3l) {
  v2f a = al(0);
  v2f b0 = b0l(0), b1 = b1l(0), b2 = b2l(0), b3 = b3l(0);
#pragma unroll 2
  for (int kk = 4; kk < K; kk += 4) {
    v2f an = al(kk);
    v2f b0n = b0l(kk), b1n = b1l(kk), b2n = b2l(kk), b3n = b3l(kk);
    WMMA4(a0, a, b0); WMMA4(a1, a, b1); WMMA4(a2, a, b2); WMMA4(a3, a, b3);
    a = an; b0 = b0n; b1 = b1n; b2 = b2n; b3 = b3n;
  }
  WMMA4(a0, a, b0); WMMA4(a1, a, b1); WMMA4(a2, a, b2); WMMA4(a3, a, b3);
}

// Load one 64x128 chunk from global, L2-normalize rows (÷ (||row||+eps)), into LDS.
static __device__ __forceinline__ void load_norm(const float* __restrict__ gsrc,
                                                 float* sM, float* sRed, float* sNrm, int t) {
  const float4* g4 = (const float4*)gsrc;
  float4* s4 = (float4*)sM;
#pragma unroll
  for (int i = 0; i < 8; ++i) s4[i * 256 + t] = g4[i * 256 + t];
  __syncthreads();
  {
    int row = t & 63, part = t >> 6;
    const float* rp = sM + row * DKk + part * 32;
    float s = 0.f;
#pragma unroll 8
    for (int c = 0; c < 32; ++c) { float x = rp[c]; s += x * x; }
    sRed[part * 64 + row] = s;
  }
  __syncthreads();
  if (t < 64) {
    float s = sRed[t] + sRed[t + 64] + sRed[t + 128] + sRed[t + 192];
    sNrm[t] = 1.0f / (sqrtf(s) + EPSN);
  }
  __syncthreads();
#pragma unroll 8
  for (int i = 0; i < 32; ++i) { int idx = i * 256 + t; sM[idx] *= sNrm[idx >> 7]; }
  __syncthreads();
}

__global__ void gdn_chunk_scan(const float* __restrict__ q,
                               const float* __restrict__ k,
                               const float* __restrict__ v,
                               const float* __restrict__ g,
                               const float* __restrict__ beta,
                               float* __restrict__ out,
                               float* __restrict__ fstate) {
  extern __shared__ float smem[];
  float* sK   = smem;               // 64x128 normalized k
  float* sQ   = sK + 8192;          // 64x128 normalized q
  float* sS   = sQ + 8192;          // 128x64 state
  float* sP   = sS + 8192;          // 64x64  A^-1 ping / correction
  float* sC   = sP + 4096;          // 64x64  A^-1 pong (final A^-1)
  float* sX0  = sC + 4096;          // 64x64  scratch
  float* sX1  = sX0 + 4096;         // 64x64  scratch
  float* sV   = sX1 + 4096;         // 64x64  raw v slice
  float* sGc  = sV + 4096;          // 64 cumsum(g)
  float* sBt  = sGc + 64;           // 64 beta
  float* sEg  = sBt + 64;           // 64 exp(cumsum g)
  float* sEgI = sEg + 64;           // 64 1/exp(cumsum g)
  float* sRed = sEgI + 64;          // 256 norm partials
  float* sNrm = sRed + 256;         // 64 inverse norms

  const int t = threadIdx.x;
  const int lane = t & 31, wave = t >> 5;
  const int lh = lane >> 4, ln = lane & 15;
  const int kofs = 2 * lh;          // K sub-offset of this lane-half

  // 64x64 tiles: wave owns row m0, columns n0 and n0+16 (A-fragment shared).
  const int m0 = (wave >> 1) * 16;
  const int n0 = (wave & 1) * 32;

  const int bh = blockIdx.x >> 2;
  const int vb = blockIdx.x & 3;

  const size_t qkBase = (size_t)bh * SEQ * DKk;
  const size_t vBase  = (size_t)bh * SEQ * DVv + (size_t)vb * DVB;
  const size_t gBase  = (size_t)bh * SEQ;

#pragma unroll
  for (int i = 0; i < 32; ++i) sS[i * 256 + t] = 0.f;
  __syncthreads();

  for (int ch = 0; ch < NCH; ++ch) {
    // ---- v slice -> sV via Tensor Data Mover (64x64 f32 tile, row stride 256) ----
#ifdef HAVE_TDM
    if (wave == 0) {
      unsigned long long ga = (unsigned long long)(const void*)(v + vBase + (size_t)ch * LQ * DVv);
      unsigned ldsOff = (unsigned)((const char*)sV - (const char*)smem);
      u32x4 g0;
      g0[0] = 1u;                                                  // count=1, user D#
      g0[1] = ldsOff;                                              // lds_addr (bytes)
      g0[2] = (unsigned)ga;                                        // global_addr[31:0]
      g0[3] = (unsigned)((ga >> 32) & 0x1FFFFFFull) | (2u << 30);  // addr[56:32] | type=2
      i32x8 g1;
      g1[0] = 2 << 16;       // data_size = 4B; workgroup_mask = 0
      g1[1] = 64 << 16;      // tensor_dim0 = 64
      g1[2] = 64 << 16;      // tensor_dim1 = 64
      g1[3] = 64 << 16;      // tile_dim0 = 64
      g1[4] = 64;            // tile_dim1 = 64, tile_dim2 = 0
      g1[5] = 256;           // tensor_dim0_stride = 256 elements
      g1[6] = 0;             // stride0 hi, stride1 lo
      g1[7] = 0;
      i32x4 z4 = {0, 0, 0, 0};
      i32x8 z8 = {0, 0, 0, 0, 0, 0, 0, 0};
      __builtin_amdgcn_tensor_load_to_lds(g0, g1, z4, z4, z8, 0);
    }
#else
    {
      const float* vg = v + vBase + (size_t)ch * LQ * DVv;
#pragma unroll
      for (int i = 0; i < 4; ++i) {
        int idx = i * 256 + t; int row = idx >> 4, c4 = idx & 15;
        ((float4*)(sV + row * DVB))[c4] = ((const float4*)(vg + (size_t)row * DVv))[c4];
      }
    }
#endif
    // ---- prefetch next chunk's q/k (global_prefetch_b8) ----
    if (ch + 1 < NCH) {
      const float* qn = q + qkBase + (size_t)(ch + 1) * LQ * DKk;
      const float* kn = k + qkBase + (size_t)(ch + 1) * LQ * DKk;
      __builtin_prefetch(qn + t * 32, 0, 3);
      __builtin_prefetch(kn + t * 32, 0, 3);
    }

    // ---- load + L2 normalize k, q ----
    load_norm(k + qkBase + (size_t)ch * LQ * DKk, sK, sRed, sNrm, t);
    load_norm(q + qkBase + (size_t)ch * LQ * DKk, sQ, sRed, sNrm, t);

    // ---- gates ----
    if (t < 64) {
      sGc[t] = g[gBase + ch * LQ + t];
      sBt[t] = beta[gBase + ch * LQ + t];
    }
    __syncthreads();
    if (t == 0) { float a = 0.f; for (int i = 0; i < 64; ++i) { a += sGc[i]; sGc[i] = a; } }
    __syncthreads();
    if (t < 64) { float e = expf(sGc[t]); sEg[t] = e; sEgI[t] = 1.0f / e; }
    __syncthreads();

    const Contig aK{sK + (m0 + ln) * DKk + kofs};
    const Contig aQ{sQ + (m0 + ln) * DKk + kofs};
    const Contig bKT0{sK + (n0 + ln) * DKk + kofs};
    const Contig bKT1{sK + (n0 + 16 + ln) * DKk + kofs};
    const Strided bS0{sS + kofs * DVB + n0 + ln, DVB};
    const Strided bS1{sS + kofs * DVB + n0 + 16 + ln, DVB};

    // ---- N[i][j] = beta_i * (k_i.k_j) * eg_j/eg_i, j < i -> sX0 ----
    {
      v8f a0 = ZACC, a1 = ZACC;
      mm2p(a0, a1, DKk, aK, bKT0, bKT1);
      int j0 = n0 + ln, j1 = n0 + 16 + ln;
      float egj0 = sEg[j0], egj1 = sEg[j1];
#pragma unroll
      for (int r = 0; r < 8; ++r) {
        int i = m0 + r + 8 * lh;
        float bi = sBt[i] * sEgI[i];
        float v0 = bi * a0[r] * egj0;
        float v1 = bi * a1[r] * egj1;
        sX0[i * LQ + j0] = (j0 < i) ? v0 : 0.f;
        sX0[i * LQ + j1] = (j1 < i) ? v1 : 0.f;
      }
    }
    __syncthreads();

    // ---- P = I - N ----
#pragma unroll 4
    for (int i = 0; i < 16; ++i) {
      int idx = i * 256 + t; int r = idx >> 6, c = idx & 63;
      sP[idx] = ((r == c) ? 1.f : 0.f) - sX0[idx];
    }
    __syncthreads();

    // ---- P <- P * (I + N^2)(I + N^4)(I + N^8)(I + N^16)(I + N^32) ----
    float* Xc = sX0; float* Xn = sX1;
    float* Pc = sP;  float* Pn = sC;
    for (int it = 0; it < 5; ++it) {
      {
        v8f a0 = ZACC, a1 = ZACC;
        mm2p(a0, a1, LQ,
             Contig{Xc + (m0 + ln) * LQ + kofs},
             Strided{Xc + kofs * LQ + n0 + ln, LQ},
             Strided{Xc + kofs * LQ + n0 + 16 + ln, LQ});
#pragma unroll
        for (int r = 0; r < 8; ++r) {
          int i = m0 + r + 8 * lh;
          Xn[i * LQ + n0 + ln] = a0[r];
          Xn[i * LQ + n0 + 16 + ln] = a1[r];
        }
      }
      __syncthreads();
      {
        v8f a0 = ZACC, a1 = ZACC;
        mm2p(a0, a1, LQ,
             Contig{Pc + (m0 + ln) * LQ + kofs},
             Strided{Xn + kofs * LQ + n0 + ln, LQ},
             Strided{Xn + kofs * LQ + n0 + 16 + ln, LQ});
#pragma unroll
        for (int r = 0; r < 8; ++r) {
          int i = m0 + r + 8 * lh;
          Pn[i * LQ + n0 + ln]      = Pc[i * LQ + n0 + ln] + a0[r];
          Pn[i * LQ + n0 + 16 + ln] = Pc[i * LQ + n0 + 16 + ln] + a1[r];
        }
      }
      __syncthreads();
      { float* tmp = Xc; Xc = Xn; Xn = tmp; }
      { float* tmp = Pc; Pc = Pn; Pn = tmp; }
    }
    float* sCorr = Pn;   // eg ⊙ correction
    float* sY = Xc;      // Y = (eg⊙k) @ S
    float* sG = Xn;      // qk*causal/eg_i*SCALE

    // ---- Y = (eg ⊙ k) @ S -> sY  (eg as output-row scale) ----
    {
      v8f a0 = ZACC, a1 = ZACC;
      mm2p(a0, a1, DKk, aK, bS0, bS1);
#pragma unroll
      for (int r = 0; r < 8; ++r) {
        int i = m0 + r + 8 * lh;
        float e = sEg[i];
        sY[i * LQ + n0 + ln] = a0[r] * e;
        sY[i * LQ + n0 + 16 + ln] = a1[r] * e;
      }
    }
#ifdef HAVE_TDM
    if (wave == 0) __builtin_amdgcn_s_wait_tensorcnt(0);  // sV ready before barrier
#endif
    __syncthreads();

    // ---- corr' = eg_i ⊙ [ P @ (beta ⊙ (eg⊙v - Y)) ] -> sCorr ----
    {
      v8f a0 = ZACC, a1 = ZACC;
      mm2p(a0, a1, LQ,
           Contig{Pc + (m0 + ln) * LQ + kofs},
           CorrB{sV + kofs * DVB + n0 + ln, sY + kofs * LQ + n0 + ln, sBt + kofs, sEg + kofs},
           CorrB{sV + kofs * DVB + n0 + 16 + ln, sY + kofs * LQ + n0 + 16 + ln, sBt + kofs, sEg + kofs});
#pragma unroll
      for (int r = 0; r < 8; ++r) {
        int i = m0 + r + 8 * lh;
        float e = sEg[i];
        sCorr[i * LQ + n0 + ln] = a0[r] * e;
        sCorr[i * LQ + n0 + 16 + ln] = a1[r] * e;
      }
    }
    __syncthreads();

    // ---- G = (q@k^T) ⊙ causal * SCALE/eg_i -> sG  (eg_j lives in corr') ----
    {
      v8f a0 = ZACC, a1 = ZACC;
      mm2p(a0, a1, DKk, aQ, bKT0, bKT1);
      int j0 = n0 + ln, j1 = n0 + 16 + ln;
#pragma unroll
      for (int r = 0; r < 8; ++r) {
        int i = m0 + r + 8 * lh;
        float gi = sEgI[i] * QSCALE;
        float v0 = a0[r] * gi;
        float v1 = a1[r] * gi;
        sG[i * LQ + j0] = (j0 <= i) ? v0 : 0.f;
        sG[i * LQ + j1] = (j1 <= i) ? v1 : 0.f;
      }
    }
    __syncthreads();

    // ---- O = SCALE*eg ⊙ (q@S) + G @ corr' ----
    {
      v8f a0 = ZACC, a1 = ZACC;
      mm2p(a0, a1, DKk, aQ, bS0, bS1);
#pragma unroll
      for (int r = 0; r < 8; ++r) {
        float sc = sEg[m0 + r + 8 * lh] * QSCALE;
        a0[r] *= sc; a1[r] *= sc;
      }
      mm2p(a0, a1, LQ,
           Contig{sG + (m0 + ln) * LQ + kofs},
           Strided{sCorr + kofs * LQ + n0 + ln, LQ},
           Strided{sCorr + kofs * LQ + n0 + 16 + ln, LQ});
      float* og = out + vBase + (size_t)ch * LQ * DVv;
#pragma unroll
      for (int r = 0; r < 8; ++r) {
        size_t ro = (size_t)(m0 + r + 8 * lh) * DVv;
        og[ro + n0 + ln] = a0[r];
        og[ro + n0 + 16 + ln] = a1[r];
      }
    }
    __syncthreads();

    // ---- S = exp(g_total) * S + k^T @ corr'  (eg folded into corr') ----
    {
      float egt = expf(sGc[63]);
      int sm0 = wave * 16;
      v8f a0 = ZACC, a1 = ZACC, a2 = ZACC, a3 = ZACC;
      mm4p(a0, a1, a2, a3, LQ,
           Strided{sK + kofs * DKk + sm0 + ln, DKk},       // k^T fragment
           Strided{sCorr + kofs * LQ + 0 + ln, LQ},
           Strided{sCorr + kofs * LQ + 16 + ln, LQ},
           Strided{sCorr + kofs * LQ + 32 + ln, LQ},
           Strided{sCorr + kofs * LQ + 48 + ln, LQ});
#pragma unroll
      for (int r = 0; r < 8; ++r) {
        int i = sm0 + r + 8 * lh;
        float* srow = sS + i * DVB;
        srow[ln]      = srow[ln] * egt + a0[r];
        srow[16 + ln] = srow[16 + ln] * egt + a1[r];
        srow[32 + ln] = srow[32 + ln] * egt + a2[r];
        srow[48 + ln] = srow[48 + ln] * egt + a3[r];
      }
    }
    __syncthreads();
  }

  // ---- final state out: [B,H,Dk,Dv] appended after output ----
#pragma unroll 4
  for (int i = 0; i < 32; ++i) {
    int idx = i * 256 + t; int row = idx >> 6, c = idx & 63;
    fstate[(size_t)bh * DKk * DVv + (size_t)row * DVv + vb * DVB + c] = sS[idx];
  }
}

#define SMEM_FLOATS (8192 * 3 + 4096 * 5 + 64 * 4 + 256 + 64)
#define SMEM_BYTES (SMEM_FLOATS * sizeof(float))

extern "C" void kernel_launch(void* const* d_in, const int* in_sizes, int n_in,
                              void* d_out, int out_size, void* d_ws, size_t ws_size,
                              hipStream_t stream) {
  (void)in_sizes; (void)n_in; (void)out_size; (void)d_ws; (void)ws_size;
  const float* q    = (const float*)d_in[0];
  const float* k    = (const float*)d_in[1];
  const float* v    = (const float*)d_in[2];
  const float* g    = (const float*)d_in[3];
  const float* beta = (const float*)d_in[4];
  float* out = (float*)d_out;                           // [B,H,S,Dv]
  float* fs  = out + (size_t)2 * 16 * 4096 * 256;       // final state [B,H,Dk,Dv]

  (void)hipFuncSetAttribute((const void*)gdn_chunk_scan,
                            hipFuncAttributeMaxDynamicSharedMemorySize,
                            (int)SMEM_BYTES);

  dim3 grid(2 * 16 * (256 / DVB));   // 128 blocks
  dim3 block(256);                   // 8 wave32s
  gdn_chunk_scan<<<grid, block, SMEM_BYTES, stream>>>(q, k, v, g, beta, out, fs);
}